// BalochiTransformer_9019431321768
// MI455X (gfx1250) — compile-verified
//
#include <hip/hip_runtime.h>
#include <hip/hip_bf16.h>

#define DIM   1024
#define HID   4096
#define SEQ   4096
#define VOCAB 128000
#define NSTEPS 10

typedef float v2f __attribute__((ext_vector_type(2)));
typedef float v8f __attribute__((ext_vector_type(8)));

struct Coef5 { float c[5]; int cnt; };
struct Coef6 { float c[6]; };

// ---------------- zero workspace accumulator ----------------
__global__ void zero_kernel(float* __restrict__ p, int n) {
    int i = blockIdx.x * blockDim.x + threadIdx.x;
    for (; i < n; i += gridDim.x * blockDim.x) p[i] = 0.0f;
}

// ---------------- embedding gather + sum (atomic) ----------------
// 32 blocks x 256 threads; block b sums tokens [b*128, b*128+128).
// thread t owns dims [t*4, t*4+4) -> per-wave contiguous 512B loads.
__global__ __launch_bounds__(256) void gather_mean_kernel(
    const int* __restrict__ tok, const float* __restrict__ emb,
    float* __restrict__ ysum) {
    int t = threadIdx.x;
    int base = blockIdx.x * 128;
    float4 acc = make_float4(0.f, 0.f, 0.f, 0.f);
    for (int s = 0; s < 128; ++s) {
        int row = tok[base + s];
        const float4* src = (const float4*)(emb + (size_t)row * DIM) + t;
        float4 v = *src;
        acc.x += v.x; acc.y += v.y; acc.z += v.z; acc.w += v.w;
    }
    atomicAdd(&ysum[t * 4 + 0], acc.x);
    atomicAdd(&ysum[t * 4 + 1], acc.y);
    atomicAdd(&ysum[t * 4 + 2], acc.z);
    atomicAdd(&ysum[t * 4 + 3], acc.w);
}

__global__ void scale_kernel(float* __restrict__ y, const float* __restrict__ ysum,
                             float inv) {
    int i = blockIdx.x * blockDim.x + threadIdx.x;
    if (i < DIM) y[i] = ysum[i] * inv;
}

// ---------------- stage A: s = w2 * (1 - tanh^2(W1 @ y_stage + b1)) --------
// y_stage = y + sum_j cf.c[j] * k_j  (computed per-block into LDS).
// One wave owns 16 rows of W1; u = W1@y_stage via V_WMMA_F32_16X16X4_F32:
//   A 16x4 tile (lanes 0-15: K=0..1, lanes 16-31: K=2..3, 2 VGPRs),
//   B 4x16 with every column = y_stage[kk..kk+3]  -> every column of D equal.
// 32 blocks x 256 threads (8 waves) -> 128 rows/block, 4096 rows total.
__global__ __launch_bounds__(256) void stageA_kernel(
    const float* __restrict__ W1, const float* __restrict__ b1,
    const float* __restrict__ w2, const float* __restrict__ y,
    const float* __restrict__ kbuf, float* __restrict__ s_out, Coef5 cf) {
    __shared__ float ys[DIM];
    int tid = threadIdx.x;
    for (int i = tid; i < DIM; i += 256) {
        float v = y[i];
        for (int j = 0; j < cf.cnt; ++j) v += cf.c[j] * kbuf[j * DIM + i];
        ys[i] = v;
    }
    __syncthreads();

    int lane = tid & 31;
    int wave = tid >> 5;
    int half = lane >> 4;        // 0: K=0..1, 1: K=2..3
    int mrow = lane & 15;        // row within 16-row tile
    int r0 = blockIdx.x * 128 + wave * 16;
    const float* arow = W1 + (size_t)(r0 + mrow) * DIM + half * 2;

    v8f c = {0.f, 0.f, 0.f, 0.f, 0.f, 0.f, 0.f, 0.f};
    for (int kk = 0; kk < DIM; kk += 4) {
        v2f a = *(const v2f*)(arow + kk);
        v2f b;
        b.x = ys[kk + half * 2];
        b.y = ys[kk + half * 2 + 1];
        c = __builtin_amdgcn_wmma_f32_16x16x4_f32(
            /*neg_a=*/false, a, /*neg_b=*/false, b,
            /*c_mod=*/(short)0, c, /*reuse_a=*/false, /*reuse_b=*/false);
    }

    // D layout: VGPR r holds M=r (lanes 0-15) and M=r+8 (lanes 16-31);
    // all columns identical, so lanes 0 and 16 carry the 16 row results.
    if ((lane & 15) == 0) {
        int rbase = r0 + (half ? 8 : 0);
#pragma unroll
        for (int r = 0; r < 8; ++r) {
            int row = rbase + r;
            float u = c[r] + b1[row];
            float t = tanhf(u);
            s_out[row] = w2[row] * (1.0f - t * t);
        }
    }
}

// ---------------- stage B: g = W1^T @ s, k_out = J g -----------------------
// 4 blocks x 256 threads; thread owns one output column. s staged through LDS
// in 256-element chunks; W1 row-segments read coalesced (L2-hot).
__global__ __launch_bounds__(256) void stageB_kernel(
    const float* __restrict__ W1, const float* __restrict__ s,
    float* __restrict__ k_out) {
    __shared__ float sl[256];
    int tid = threadIdx.x;
    int col = blockIdx.x * 256 + tid;
    float acc = 0.0f;
    for (int c0 = 0; c0 < HID; c0 += 256) {
        sl[tid] = s[c0 + tid];
        __syncthreads();
#pragma unroll 8
        for (int ii = 0; ii < 256; ++ii)
            acc = fmaf(W1[(size_t)(c0 + ii) * DIM + col], sl[ii], acc);
        __syncthreads();
    }
    // field = [g_p, -g_q]: col<512 contributes -g to index col+512,
    //                      col>=512 contributes +g to index col-512.
    int outi = (col < 512) ? (col + 512) : (col - 512);
    k_out[outi] = (col < 512) ? -acc : acc;
}

// ---------------- per-step update: y += sum_j h*b_j * k_j ------------------
__global__ void update_kernel(float* __restrict__ y, const float* __restrict__ kbuf,
                              Coef6 cf) {
    int i = blockIdx.x * blockDim.x + threadIdx.x;
    if (i < DIM) {
        float v = y[i];
#pragma unroll
        for (int j = 0; j < 6; ++j) v += cf.c[j] * kbuf[j * DIM + i];
        y[i] = v;
    }
}

// ---------------- decoder: out = dec_w @ y + dec_b -------------------------
// 512 MB streamed from HBM -> bandwidth-bound; wave-per-row, b128 loads,
// y in LDS, shuffle reduction. Prefetch next block's rows.
__global__ __launch_bounds__(256) void decode_kernel(
    const float* __restrict__ dec_w, const float* __restrict__ dec_b,
    const float* __restrict__ y, float* __restrict__ out) {
    __shared__ float yl[DIM];
    int tid = threadIdx.x;
#pragma unroll
    for (int j = 0; j < 4; ++j) yl[tid + 256 * j] = y[tid + 256 * j];
    __syncthreads();

    int lane = tid & 31;
    int wave = tid >> 5;
    int row = blockIdx.x * 8 + wave;
    const float* wrow = dec_w + (size_t)row * DIM;
    __builtin_prefetch(wrow + 8 * DIM, 0, 0);  // next block's row group

    float acc = 0.0f;
#pragma unroll
    for (int it = 0; it < 8; ++it) {
        int idx = it * 128 + lane * 4;
        float4 w = *(const float4*)(wrow + idx);
        acc = fmaf(w.x, yl[idx + 0], acc);
        acc = fmaf(w.y, yl[idx + 1], acc);
        acc = fmaf(w.z, yl[idx + 2], acc);
        acc = fmaf(w.w, yl[idx + 3], acc);
    }
#pragma unroll
    for (int off = 16; off > 0; off >>= 1) acc += __shfl_xor(acc, off, 32);
    if (lane == 0) out[row] = acc + dec_b[row];
}

// ---------------------------------------------------------------------------
extern "C" void kernel_launch(void* const* d_in, const int* in_sizes, int n_in,
                              void* d_out, int out_size, void* d_ws, size_t ws_size,
                              hipStream_t stream) {
    (void)in_sizes; (void)n_in; (void)out_size; (void)ws_size;
    const int*   tok   = (const int*)d_in[0];
    const float* emb   = (const float*)d_in[1];
    const float* W1    = (const float*)d_in[2];
    const float* b1    = (const float*)d_in[3];
    const float* w2    = (const float*)d_in[4];
    /* d_in[5] = b2 scalar: additive constant in H, vanishes in grad */
    const float* dec_w = (const float*)d_in[6];
    const float* dec_b = (const float*)d_in[7];
    float* out = (float*)d_out;

    float* ws   = (float*)d_ws;
    float* y    = ws;                      // 1024
    float* ysum = ws + 1024;               // 1024
    float* kbuf = ws + 2048;               // 6*1024
    float* sbuf = ws + 2048 + 6 * 1024;    // 4096

    // Tsit5 tableau
    static const float Atab[5][5] = {
        {0.161f, 0.f, 0.f, 0.f, 0.f},
        {-0.008480655492356989f, 0.335480655492357f, 0.f, 0.f, 0.f},
        {2.8971530571054935f, -6.359448489975075f, 4.3622954328695815f, 0.f, 0.f},
        {5.325864828439257f, -11.748883564062828f, 7.4955393428898365f,
         -0.09249506636175525f, 0.f},
        {5.86145544294642f, -12.92096931784711f, 8.159367898576159f,
         -0.071584973281401f, -0.028269050394068383f}};
    static const float Btab[6] = {0.09646076681806523f, 0.01f, 0.4798896504144996f,
                                  1.379008574103742f, -3.290069515436081f,
                                  2.324710524099774f};
    const float h = 0.1f;

    zero_kernel<<<4, 256, 0, stream>>>(ysum, DIM);
    gather_mean_kernel<<<32, 256, 0, stream>>>(tok, emb, ysum);
    scale_kernel<<<4, 256, 0, stream>>>(y, ysum, 1.0f / (float)SEQ);

    for (int step = 0; step < NSTEPS; ++step) {
        for (int st = 0; st < 6; ++st) {
            Coef5 cf{};
            cf.cnt = st;
            for (int j = 0; j < st; ++j) cf.c[j] = h * Atab[st - 1][j];
            stageA_kernel<<<32, 256, 0, stream>>>(W1, b1, w2, y, kbuf, sbuf, cf);
            stageB_kernel<<<4, 256, 0, stream>>>(W1, sbuf, kbuf + st * DIM);
        }
        Coef6 cu{};
        for (int j = 0; j < 6; ++j) cu.c[j] = h * Btab[j];
        update_kernel<<<4, 256, 0, stream>>>(y, kbuf, cu);
    }

    decode_kernel<<<VOCAB / 8, 256, 0, stream>>>(dec_w, dec_b, y, out);
}